// GraphDecoderEVRP_60498909331777
// MI455X (gfx1250) — compile-verified
//
#include <hip/hip_runtime.h>

// ---------------------------------------------------------------------------
// GraphDecoderEVRP decode step, MI455X (gfx1250), wave32.
// Bandwidth-bound: streams glimpse_K/glimpse_V/logit_K (1.57 GB) once.
// WMMA (V_WMMA_F32_16X16X4_F32) used for the two dense projections.
// ---------------------------------------------------------------------------

typedef float v2f __attribute__((ext_vector_type(2)));
typedef float f4  __attribute__((ext_vector_type(4)));
typedef float v8f __attribute__((ext_vector_type(8)));

#define Bn  1024
#define Nn  1000
#define En  128
#define Hn  8
#define DKn 16
#define SCn 256

static __device__ __forceinline__ f4 ntload(const f4* p) {
  return __builtin_nontemporal_load(p);
}
static __device__ __forceinline__ float dot4(f4 a, f4 b) {
  return a.x * b.x + a.y * b.y + a.z * b.z + a.w * b.w;
}

// ---------------------------------------------------------------------------
// Kernel 1: query[b,e] = context[b,e] + sum_c step_context[b,c] * W_step[e,c]
// GEMM 1024x128 (K=256) via V_WMMA_F32_16X16X4_F32; 1 wave per 16x16 tile.
// f32 A 16x4 layout: lane(l) row M=l&15; VGPR0 holds K=2*(l>>4), VGPR1 K=+1.
// f32 C/D 16x16  : elem r of lane l -> (M = r + 8*(l>>4), N = l&15).
// ---------------------------------------------------------------------------
__global__ __launch_bounds__(256) void k_query(const float* __restrict__ ctx,
                                               const float* __restrict__ sc,
                                               const float* __restrict__ Wstep,
                                               float* __restrict__ query) {
  const int lane = threadIdx.x & 31;
  const int wave = threadIdx.x >> 5;
  const int tile = blockIdx.x * 8 + wave;  // 512 tiles = 64 b-tiles x 8 e-tiles
  const int b0 = (tile >> 3) * 16;
  const int e0 = (tile & 7) * 16;
  const int half = lane >> 4;
  const int l16  = lane & 15;

  v8f c;
#pragma unroll
  for (int r = 0; r < 8; ++r)
    c[r] = ctx[(b0 + r + 8 * half) * En + e0 + l16];  // D = A*B + context

  const float* arow = sc    + (b0 + l16) * SCn + 2 * half;
  const float* brow = Wstep + (e0 + l16) * SCn + 2 * half;  // B[k,e]=W_step[e,k]
#pragma unroll 4
  for (int kk = 0; kk < SCn; kk += 4) {
    v2f a  = *(const v2f*)(arow + kk);
    v2f bb = *(const v2f*)(brow + kk);
    c = __builtin_amdgcn_wmma_f32_16x16x4_f32(false, a, false, bb, (short)0, c,
                                              false, false);
  }
#pragma unroll
  for (int r = 0; r < 8; ++r)
    query[(b0 + r + 8 * half) * En + e0 + l16] = c[r];
}

// ---------------------------------------------------------------------------
// Kernel 2: fused masked attention for one (h,b) per block.
// compat = K·q/4 -> masked softmax -> heads = attn·V -> ws glimpse[b, h*16+k]
// Streams 1 GB total with nontemporal b128 loads.
// ---------------------------------------------------------------------------
__global__ __launch_bounds__(256) void k_attn(const float* __restrict__ gK,
                                              const float* __restrict__ gV,
                                              const unsigned char* __restrict__ mask,
                                              const float* __restrict__ query,
                                              float* __restrict__ glimpse) {
  __shared__ __align__(16) float qs[DKn];
  __shared__ float probs[Nn];
  __shared__ float red[256];
  __shared__ float wpart[8][DKn];

  const int tid = threadIdx.x;
  const int bid = blockIdx.x;     // h*1024 + b  (matches glimpse_K [H,B,N,dk])
  const int h = bid >> 10;
  const int b = bid & 1023;
  const float* Kbase = gK + (size_t)bid * Nn * DKn;
  const float* Vbase = gV + (size_t)bid * Nn * DKn;
  const unsigned char* mrow = mask + (size_t)b * Nn;

  if (tid < DKn) qs[tid] = query[b * En + h * DKn + tid] * 0.25f;  // 1/sqrt(16)
  __syncthreads();
  const f4 q0 = *(const f4*)&qs[0];
  const f4 q1 = *(const f4*)&qs[4];
  const f4 q2 = *(const f4*)&qs[8];
  const f4 q3 = *(const f4*)&qs[12];

  float lmax = -INFINITY;
  for (int n = tid; n < Nn; n += 256) {
    const f4* kr = (const f4*)(Kbase + (size_t)n * DKn);
    float cv = dot4(ntload(kr + 0), q0) + dot4(ntload(kr + 1), q1) +
               dot4(ntload(kr + 2), q2) + dot4(ntload(kr + 3), q3);
    if (mrow[n]) cv = -INFINITY;
    probs[n] = cv;
    lmax = fmaxf(lmax, cv);
  }
  red[tid] = lmax;
  __syncthreads();
  for (int s = 128; s > 0; s >>= 1) {
    if (tid < s) red[tid] = fmaxf(red[tid], red[tid + s]);
    __syncthreads();
  }
  const float gmax = red[0];
  __syncthreads();

  float lsum = 0.f;
  for (int n = tid; n < Nn; n += 256) {
    float e = __expf(probs[n] - gmax);  // exp(-inf - finite) = 0
    probs[n] = e;
    lsum += e;
  }
  red[tid] = lsum;
  __syncthreads();
  for (int s = 128; s > 0; s >>= 1) {
    if (tid < s) red[tid] += red[tid + s];
    __syncthreads();
  }
  const float inv = 1.0f / red[0];

  float acc[DKn];
#pragma unroll
  for (int k = 0; k < DKn; ++k) acc[k] = 0.f;
  for (int n = tid; n < Nn; n += 256) {
    const float a = probs[n];
    const f4* vr = (const f4*)(Vbase + (size_t)n * DKn);
    const f4 v0 = ntload(vr + 0), v1 = ntload(vr + 1);
    const f4 v2 = ntload(vr + 2), v3 = ntload(vr + 3);
    acc[0] += a * v0.x;  acc[1] += a * v0.y;  acc[2]  += a * v0.z;  acc[3]  += a * v0.w;
    acc[4] += a * v1.x;  acc[5] += a * v1.y;  acc[6]  += a * v1.z;  acc[7]  += a * v1.w;
    acc[8] += a * v2.x;  acc[9] += a * v2.y;  acc[10] += a * v2.z;  acc[11] += a * v2.w;
    acc[12]+= a * v3.x;  acc[13]+= a * v3.y;  acc[14] += a * v3.z;  acc[15] += a * v3.w;
  }
#pragma unroll
  for (int k = 0; k < DKn; ++k)
    for (int off = 16; off > 0; off >>= 1)
      acc[k] += __shfl_down(acc[k], off, 32);
  if ((tid & 31) == 0)
#pragma unroll
    for (int k = 0; k < DKn; ++k) wpart[tid >> 5][k] = acc[k];
  __syncthreads();
  if (tid < DKn) {
    float s = 0.f;
#pragma unroll
    for (int w = 0; w < 8; ++w) s += wpart[w][tid];
    glimpse[b * En + h * DKn + tid] = s * inv;  // concat heads: [B, H*dk]
  }
}

// ---------------------------------------------------------------------------
// Kernel 3: glimpse_proj[b,o] = sum_e glimpse[b,e] * W_out[o,e]
// GEMM 1024x128 (K=128) via V_WMMA_F32_16X16X4_F32.
// ---------------------------------------------------------------------------
__global__ __launch_bounds__(256) void k_proj(const float* __restrict__ gl,
                                              const float* __restrict__ Wout,
                                              float* __restrict__ gp) {
  const int lane = threadIdx.x & 31;
  const int wave = threadIdx.x >> 5;
  const int tile = blockIdx.x * 8 + wave;
  const int b0 = (tile >> 3) * 16;
  const int o0 = (tile & 7) * 16;
  const int half = lane >> 4;
  const int l16  = lane & 15;

  v8f c = {};
  const float* arow = gl   + (b0 + l16) * En + 2 * half;
  const float* brow = Wout + (o0 + l16) * En + 2 * half;  // B[e,o]=W_out[o,e]
#pragma unroll 4
  for (int kk = 0; kk < En; kk += 4) {
    v2f a  = *(const v2f*)(arow + kk);
    v2f bb = *(const v2f*)(brow + kk);
    c = __builtin_amdgcn_wmma_f32_16x16x4_f32(false, a, false, bb, (short)0, c,
                                              false, false);
  }
#pragma unroll
  for (int r = 0; r < 8; ++r)
    gp[(b0 + r + 8 * half) * En + o0 + l16] = c[r];
}

// ---------------------------------------------------------------------------
// Kernel 4: logits[b,n] = tanh(logit_K[b,n,:]·gp[b,:]/sqrt(128))*10, mask,
// log_softmax over n. One block per b; streams 512 MB nontemporal.
// ---------------------------------------------------------------------------
__global__ __launch_bounds__(256) void k_logits(const float* __restrict__ lK,
                                                const float* __restrict__ gp,
                                                const unsigned char* __restrict__ mask,
                                                float* __restrict__ out) {
  __shared__ __align__(16) float gs[En];
  __shared__ float lg[Nn];
  __shared__ float red[256];
  const int tid = threadIdx.x;
  const int b = blockIdx.x;
  if (tid < En) gs[tid] = gp[b * En + tid];
  __syncthreads();

  const float* Lbase = lK + (size_t)b * Nn * En;
  const unsigned char* mrow = mask + (size_t)b * Nn;
  float lmax = -INFINITY;
  for (int n = tid; n < Nn; n += 256) {
    const f4* r = (const f4*)(Lbase + (size_t)n * En);
    float s = 0.f;
#pragma unroll 8
    for (int k = 0; k < En / 4; ++k) s += dot4(ntload(r + k), *(const f4*)&gs[4 * k]);
    s = tanhf(s * 0.08838834764831843f) * 10.0f;  // /sqrt(128), tanh clip
    if (mrow[n]) s = -INFINITY;
    lg[n] = s;
    lmax = fmaxf(lmax, s);
  }
  red[tid] = lmax;
  __syncthreads();
  for (int s = 128; s > 0; s >>= 1) {
    if (tid < s) red[tid] = fmaxf(red[tid], red[tid + s]);
    __syncthreads();
  }
  const float gmax = red[0];
  __syncthreads();
  float lsum = 0.f;
  for (int n = tid; n < Nn; n += 256) lsum += __expf(lg[n] - gmax);
  red[tid] = lsum;
  __syncthreads();
  for (int s = 128; s > 0; s >>= 1) {
    if (tid < s) red[tid] += red[tid + s];
    __syncthreads();
  }
  const float lse = gmax + __logf(red[0]);
  for (int n = tid; n < Nn; n += 256) out[(size_t)b * Nn + n] = lg[n] - lse;
}

// ---------------------------------------------------------------------------
extern "C" void kernel_launch(void* const* d_in, const int* in_sizes, int n_in,
                              void* d_out, int out_size, void* d_ws, size_t ws_size,
                              hipStream_t stream) {
  (void)in_sizes; (void)n_in; (void)out_size; (void)ws_size;
  const float* ctx   = (const float*)d_in[0];  // [B,1,E]
  const float* sc    = (const float*)d_in[1];  // [B,1,SC]
  const float* Wstep = (const float*)d_in[2];  // [E,SC]
  const float* gK    = (const float*)d_in[3];  // [H,B,1,N,dk]
  const float* gV    = (const float*)d_in[4];  // [H,B,1,N,dk]
  const float* lK    = (const float*)d_in[5];  // [B,1,N,E]
  const float* Wout  = (const float*)d_in[6];  // [E,E]
  const unsigned char* mask = (const unsigned char*)d_in[7];  // [B,1,N] bool
  float* out = (float*)d_out;                  // [B,1,N]

  float* q  = (float*)d_ws;      // [B,E]  query
  float* gl = q + Bn * En;       // [B,E]  concat heads
  float* gp = gl + Bn * En;      // [B,E]  projected glimpse

  k_query <<<64,        256, 0, stream>>>(ctx, sc, Wstep, q);
  k_attn  <<<Hn * Bn,   256, 0, stream>>>(gK, gV, mask, q, gl);
  k_proj  <<<64,        256, 0, stream>>>(gl, Wout, gp);
  k_logits<<<Bn,        256, 0, stream>>>(lK, gp, mask, out);
}